// PostProcessingNetwork_30030411334376
// MI455X (gfx1250) — compile-verified
//
#include <hip/hip_runtime.h>
#include <math.h>

#define L 2048
#define LSHIFT 11
#define T_STEPS 20
#define STRIPS (L / 16)       // 128 row strips
#define SEGS 8                // waves per block; each wave covers L/SEGS columns
#define SEG_COLS (L / SEGS)   // 256

typedef float v2f __attribute__((ext_vector_type(2)));
typedef float v8f __attribute__((ext_vector_type(8)));

__device__ __forceinline__ float sigmoidf(float x) {
    return 1.0f / (1.0f + expf(-x));
}

// ---------------------------------------------------------------------------
// Prep: S = 0.5*(U_proc + U_proc^T), A_hat0 = tanh(U-s)*sigmoid(U)
// U_proc = tanh(U-s)*U, computed on the fly (no temp buffer).
// ---------------------------------------------------------------------------
__global__ void prep_kernel(const float* __restrict__ U,
                            const float* __restrict__ sP,
                            float* __restrict__ S,
                            float* __restrict__ Ah) {
    int idx = blockIdx.x * blockDim.x + threadIdx.x;
    int i = idx >> LSHIFT;
    int j = idx & (L - 1);
    float s  = sP[0];
    float u  = U[idx];
    float ut = U[(size_t)j * L + i];
    float th = tanhf(u - s);
    float up  = th * u;
    float upt = tanhf(ut - s) * ut;
    S[idx]  = 0.5f * (up + upt);
    Ah[idx] = th * sigmoidf(u);
}

// ---------------------------------------------------------------------------
// Elementwise A_hat update (in place):
//   A_dot = ah*(1 + alpha*da*M*(S - c[i] - c[j]))
//   ah'   = clip(relu(|A_dot| - rho*alpha*da), 0, 1)
// ---------------------------------------------------------------------------
__global__ void update_kernel(float* __restrict__ Ah,
                              const float* __restrict__ S,
                              const float* __restrict__ M,
                              const float* __restrict__ c,
                              const float* __restrict__ alphaP,
                              const float* __restrict__ gaP,
                              const float* __restrict__ rhoP,
                              int t) {
    int idx = blockIdx.x * blockDim.x + threadIdx.x;
    int i = idx >> LSHIFT;
    int j = idx & (L - 1);
    float ada = alphaP[0] * powf(gaP[0], (float)t);
    float thr = rhoP[0] * ada;
    float ah  = Ah[idx];
    float g   = S[idx] - c[i] - c[j];
    float ad  = ah + ada * ah * M[idx] * g;
    float v   = fabsf(ad) - thr;
    v = fminf(fmaxf(v, 0.0f), 1.0f);
    Ah[idx] = v;
}

// ---------------------------------------------------------------------------
// Transform + WMMA row-sum:
//   A[i][j] = 0.5*(ah[i][j]^2 + ah[j][i]^2)*M[i][j]
// Each wave owns a 16-row strip segment; values are produced directly in the
// V_WMMA_F32_16X16X4_F32 A-operand layout (lane<16: M=lane, K={0,1};
// lane>=16: M=lane-16, K={2,3}), multiplied by all-ones B so the accumulator
// carries row sums. WRITE_OUT is a compile-time flag so the store path has no
// EXEC masking and EXEC is statically all-ones around every WMMA. Trajectory
// is written with non-temporal stores so the 335MB output stream does not
// evict the L2-resident working set (S, M, A_hat ~ 48MB << 192MB L2).
// ---------------------------------------------------------------------------
template <bool WRITE_OUT>
__global__ void transform_kernel(const float* __restrict__ Ah,
                                 const float* __restrict__ M,
                                 float* __restrict__ out,        // used iff WRITE_OUT
                                 float* __restrict__ partials) { // [STRIPS][SEGS][16]
    int strip = blockIdx.x;
    int wid = threadIdx.x >> 5;   // wave id in block: 0..SEGS-1
    int lid = threadIdx.x & 31;
    int half = lid >> 4;          // 0 -> K {0,1}, 1 -> K {2,3}
    int m = lid & 15;
    int i = strip * 16 + m;

    const float* ahRow = Ah + (size_t)i * L;
    const float* mRow  = M  + (size_t)i * L;
    float* oRow = WRITE_OUT ? (out + (size_t)i * L) : nullptr;

    v8f acc = {0.0f, 0.0f, 0.0f, 0.0f, 0.0f, 0.0f, 0.0f, 0.0f};
    v2f ones;
    ones.x = 1.0f;
    ones.y = 1.0f;

    int cbase = wid * SEG_COLS;
    for (int c0 = cbase; c0 < cbase + SEG_COLS; c0 += 4) {
        int j0 = c0 + half * 2;
        v2f aij = *(const v2f*)(ahRow + j0);         // ah[i][j0], ah[i][j0+1]
        v2f mij = *(const v2f*)(mRow + j0);
        float aj0 = Ah[(size_t)j0 * L + i];          // ah[j0][i]   (coalesced)
        float aj1 = Ah[(size_t)(j0 + 1) * L + i];    // ah[j0+1][i] (coalesced)
        v2f a;
        a.x = 0.5f * (aij.x * aij.x + aj0 * aj0) * mij.x;
        a.y = 0.5f * (aij.y * aij.y + aj1 * aj1) * mij.y;
        if (WRITE_OUT) {
            __builtin_nontemporal_store(a, (v2f*)(oRow + j0));
        }
        // D = A(16x4 data) * B(4x16 ones) + C  ->  row sums accumulate in C
        acc = __builtin_amdgcn_wmma_f32_16x16x4_f32(
            false, a, false, ones, (short)0, acc, false, false);
    }

    // D layout: lane 0 holds D[m=k][0] in acc[k] (rows 0..7);
    //           lane 16 holds D[m=k+8][0] in acc[k] (rows 8..15).
    float* p = partials + ((size_t)strip * SEGS + wid) * 16;
    if (lid == 0) {
#pragma unroll
        for (int k = 0; k < 8; ++k) p[k] = acc[k];
    } else if (lid == 16) {
#pragma unroll
        for (int k = 0; k < 8; ++k) p[8 + k] = acc[k];
    }
}

// ---------------------------------------------------------------------------
// Reduce per-strip partial row sums; update lam and c.
//   init: lam = w * relu(rs)
//   step: lam += beta * gb^t * relu(rs)
//   c = lam * sigmoid(10*rs)
// ---------------------------------------------------------------------------
__global__ void tail_kernel(const float* __restrict__ partials,
                            float* __restrict__ lam,
                            float* __restrict__ c,
                            const float* __restrict__ coefP, // w (init) or beta
                            const float* __restrict__ gbP,
                            int t, int init) {
    int i = blockIdx.x * blockDim.x + threadIdx.x; // 0..L-1
    int strip = i >> 4;
    int r = i & 15;
    float rsum = 0.0f;
#pragma unroll
    for (int seg = 0; seg < SEGS; ++seg)
        rsum += partials[((size_t)strip * SEGS + seg) * 16 + r];
    float rs = rsum - 1.0f;
    float rel = fmaxf(rs, 0.0f);
    float lv;
    if (init) {
        lv = coefP[0] * rel;
    } else {
        lv = lam[i] + coefP[0] * powf(gbP[0], (float)t) * rel;
    }
    lam[i] = lv;
    c[i] = lv * sigmoidf(10.0f * rs);
}

// ---------------------------------------------------------------------------
extern "C" void kernel_launch(void* const* d_in, const int* in_sizes, int n_in,
                              void* d_out, int out_size, void* d_ws, size_t ws_size,
                              hipStream_t stream) {
    const float* U      = (const float*)d_in[0];
    const float* M      = (const float*)d_in[1];
    const float* sP     = (const float*)d_in[2];
    const float* alphaP = (const float*)d_in[3];
    const float* betaP  = (const float*)d_in[4];
    const float* gaP    = (const float*)d_in[5];
    const float* gbP    = (const float*)d_in[6];
    const float* rhoP   = (const float*)d_in[7];
    const float* wP     = (const float*)d_in[8];
    float* out = (float*)d_out;

    float* S        = (float*)d_ws;                          // L*L
    float* Ah       = S + (size_t)L * L;                     // L*L
    float* partials = Ah + (size_t)L * L;                    // STRIPS*SEGS*16
    float* lam      = partials + (size_t)STRIPS * SEGS * 16; // L
    float* c        = lam + L;                               // L

    dim3 blk(256);
    const int nElemBlocks = (L * L) / 256;

    prep_kernel<<<nElemBlocks, blk, 0, stream>>>(U, sP, S, Ah);
    transform_kernel<false><<<STRIPS, blk, 0, stream>>>(Ah, M, nullptr, partials);
    tail_kernel<<<L / 256, blk, 0, stream>>>(partials, lam, c, wP, betaP, 0, 1);

    for (int t = 0; t < T_STEPS; ++t) {
        update_kernel<<<nElemBlocks, blk, 0, stream>>>(Ah, S, M, c,
                                                       alphaP, gaP, rhoP, t);
        transform_kernel<true><<<STRIPS, blk, 0, stream>>>(
            Ah, M, out + (size_t)t * L * L, partials);
        tail_kernel<<<L / 256, blk, 0, stream>>>(partials, lam, c,
                                                 betaP, gbP, t, 0);
    }
}